// EuclideanAttentionBlock_13889924235381
// MI455X (gfx1250) — compile-verified
//
#include <hip/hip_runtime.h>

typedef float v2f __attribute__((ext_vector_type(2)));
typedef float v4f __attribute__((ext_vector_type(4)));
typedef float v8f __attribute__((ext_vector_type(8)));

#define N_NODES 10000
#define N_EDGES 160000
#define FDIM    240
#define RBF_DIM 32
#define EV_DIM  15
#define KIN     35          // RBF + 3 degree invariants
#define KPAD    36          // padded K (slot 35 carries the bias, f_in[35]=1)
#define WT_SZ   (FDIM * KPAD)   // 8640 floats per transposed filter

// --------------------------------------------------------------------------
__global__ void zero_kernel(float* __restrict__ p, int n) {
    int i = blockIdx.x * blockDim.x + threadIdx.x;
    if (i < n) p[i] = 0.0f;
}

// --------------------------------------------------------------------------
// per-edge SO3 invariants: sum over m of (ev[s]-ev[r])^2 per degree -> [E,3]
__global__ void edge_invs_kernel(const float* __restrict__ ev,
                                 const int* __restrict__ senders,
                                 const int* __restrict__ receivers,
                                 float* __restrict__ invs) {
    int t = blockIdx.x * blockDim.x + threadIdx.x;
    if (t >= N_EDGES) return;
    int s = senders[t], r = receivers[t];
    float a0 = 0.f, a1 = 0.f, a2 = 0.f;
#pragma unroll
    for (int j = 0; j < EV_DIM; ++j) {
        float d  = ev[s * EV_DIM + j] - ev[r * EV_DIM + j];
        float d2 = d * d;
        if (j < 3)      a0 += d2;
        else if (j < 8) a1 += d2;
        else            a2 += d2;
    }
    invs[t * 3 + 0] = a0;
    invs[t * 3 + 1] = a1;
    invs[t * 3 + 2] = a2;
}

// --------------------------------------------------------------------------
// Filter weights: W[35,240] (+bias[240]) -> Wt[240,36]; slot k=35 = bias.
__global__ void transpose_filt_kernel(const float* __restrict__ W,
                                      const float* __restrict__ b,
                                      float* __restrict__ Wt) {
    int i = blockIdx.x * blockDim.x + threadIdx.x;
    if (i >= WT_SZ) return;
    int m = i / KPAD, k = i % KPAD;
    Wt[i] = (k < KIN) ? W[k * FDIM + m] : b[m];
}

// Projection: W[h][D][D] (i,j) -> Wt[h][JP][D] (j,k), rows j>=D zeroed.
template <int D, int JP>
__global__ void transpose_proj_kernel(const float* __restrict__ W,
                                      float* __restrict__ Wt, int total) {
    int i = blockIdx.x * blockDim.x + threadIdx.x;
    if (i >= total) return;
    int h = i / (JP * D), rem = i % (JP * D);
    int j = rem / D, k = rem % D;
    Wt[i] = (j < D) ? W[h * D * D + k * D + j] : 0.f;
}

// --------------------------------------------------------------------------
// Per-head node projections: Y[n,h,j] = sum_i X[n,h,i] * W[h,i,j]
// One wave per (16-node tile, head, 16-col tile). All fragments preloaded
// into arrays -> one load clause, then a pure back-to-back WMMA chain.
template <int D, int JP, int NMAT>
__global__ __launch_bounds__(32)
void proj_kernel(const float* __restrict__ X,
                 const float* __restrict__ Wt0,
                 const float* __restrict__ Wt1,
                 const float* __restrict__ Wt2,
                 float* __restrict__ Y0,
                 float* __restrict__ Y1,
                 float* __restrict__ Y2) {
    constexpr int KS = D / 4;
    const int lane = threadIdx.x;
    const int half = lane >> 4;
    const int ln   = lane & 15;
    const int ntile = blockIdx.x;
    const int head  = blockIdx.y;
    const int colbase = blockIdx.z * 16;
    const int node = ntile * 16 + ln;     // A row this lane supplies
    const int j    = colbase + ln;        // B/D column this lane supplies (j<JP)

    const float* w0 = Wt0 + (head * JP + j) * D;
    const float* w1 = Wt1 + (head * JP + j) * D;
    const float* w2 = Wt2 + (head * JP + j) * D;
    const float* xr = X + node * FDIM + head * D;

    v2f a[KS], b0[KS], b1[KS], b2[KS];
#pragma unroll
    for (int ks = 0; ks < KS; ++ks) {
        const int k0 = ks * 4 + 2 * half;          // K = vreg + 2*half
        a[ks]  = *(const v2f*)&xr[k0];
        b0[ks] = *(const v2f*)&w0[k0];
        if (NMAT > 1) b1[ks] = *(const v2f*)&w1[k0];
        if (NMAT > 2) b2[ks] = *(const v2f*)&w2[k0];
    }

    v8f c0 = {0, 0, 0, 0, 0, 0, 0, 0};
    v8f c1 = {0, 0, 0, 0, 0, 0, 0, 0};
    v8f c2 = {0, 0, 0, 0, 0, 0, 0, 0};
#pragma unroll
    for (int ks = 0; ks < KS; ++ks) {
        c0 = __builtin_amdgcn_wmma_f32_16x16x4_f32(
            false, a[ks], false, b0[ks], (short)0, c0, false, false);
        if (NMAT > 1)
            c1 = __builtin_amdgcn_wmma_f32_16x16x4_f32(
                false, a[ks], false, b1[ks], (short)0, c1, false, false);
        if (NMAT > 2)
            c2 = __builtin_amdgcn_wmma_f32_16x16x4_f32(
                false, a[ks], false, b2[ks], (short)0, c2, false, false);
    }

    if (j < D) {
#pragma unroll
        for (int r = 0; r < 8; ++r) {              // D row = r + 8*half
            const int nd = ntile * 16 + r + 8 * half;
            Y0[nd * FDIM + head * D + j] = c0[r];
            if (NMAT > 1) Y1[nd * FDIM + head * D + j] = c1[r];
            if (NMAT > 2) Y2[nd * FDIM + head * D + j] = c2[r];
        }
    }
}

// --------------------------------------------------------------------------
// Fused edge kernel: fw^T = W_filt^T @ f_in^T (M=feature, N=edge); each
// lane's D-column is its own edge. Both transposed filter matrices are
// staged in LDS (69 KB, conflict-free ds_load_b64), fragments preloaded
// into arrays per tile, outputs scattered with fp32 atomics.
__global__ __launch_bounds__(128)
void edge_attn_kernel(const float* __restrict__ rbf,
                      const float* __restrict__ sh,
                      const float* __restrict__ cutoffs,
                      const float* __restrict__ wtf,   // [2][240][36] inv,ev
                      const int* __restrict__ senders,
                      const int* __restrict__ receivers,
                      const float* __restrict__ qi, const float* __restrict__ ki,
                      const float* __restrict__ vi,
                      const float* __restrict__ qe, const float* __restrict__ ke,
                      const float* __restrict__ invs,
                      float* __restrict__ d_inv, float* __restrict__ d_ev) {
    extern __shared__ float ldsw[];   // 2 * 240 * 36 floats = 69120 B

    // cooperative stage of both filter matrices into LDS
    for (int idx = threadIdx.x; idx < (2 * WT_SZ) / 4; idx += 128)
        ((v4f*)ldsw)[idx] = ((const v4f*)wtf)[idx];
    __syncthreads();

    const int wave = threadIdx.x >> 5;
    const int lane = threadIdx.x & 31;
    const int half = lane >> 4;
    const int ln   = lane & 15;
    const int tile = blockIdx.x * 4 + wave;   // 10000 tiles (160000/16 exact)
    const int e    = tile * 16 + ln;          // this lane's edge

    const int snd = senders[e];
    const int rcv = receivers[e];
    const float cut = cutoffs[e];

    // B fragments of f_in^T (K = vreg + 2*half), loaded once, reused 30x
    v2f bfr[9];
#pragma unroll
    for (int ks = 0; ks < 8; ++ks)
        bfr[ks] = *(const v2f*)&rbf[e * RBF_DIM + ks * 4 + 2 * half];
    {
        float iy  = invs[e * 3 + 1];
        bfr[8].x  = invs[e * 3 + 2 * half];
        bfr[8].y  = half ? 1.0f : iy;           // K=35 slot: bias multiplier
    }

    float ai0 = 0.f, ai1 = 0.f, ai2 = 0.f, ai3 = 0.f;  // 4 inv heads
    float ae0 = 0.f, ae1 = 0.f, ae2 = 0.f;             // 3 ev heads

    const float* qir = qi + rcv * FDIM;
    const float* kir = ki + snd * FDIM;
    const float* qer = qe + rcv * FDIM;
    const float* ker = ke + snd * FDIM;

#pragma unroll
    for (int mt = 0; mt < 15; ++mt) {
        const float* wri = ldsw + (mt * 16 + ln) * KPAD;          // inv filter
        const float* wre = wri + WT_SZ;                            // ev filter
        const int fbase = mt * 16 + 8 * half;   // first feature this lane holds

        // preload all fragments + consume operands (independent registers)
        v2f afi[9], afe[9];
#pragma unroll
        for (int ks = 0; ks < 9; ++ks) {
            const int k0 = ks * 4 + 2 * half;
            afi[ks] = *(const v2f*)&wri[k0];
            afe[ks] = *(const v2f*)&wre[k0];
        }
        v4f qi0 = *(const v4f*)&qir[fbase], qi1 = *(const v4f*)&qir[fbase + 4];
        v4f ki0 = *(const v4f*)&kir[fbase], ki1 = *(const v4f*)&kir[fbase + 4];
        v4f qe0 = *(const v4f*)&qer[fbase], qe1 = *(const v4f*)&qer[fbase + 4];
        v4f ke0 = *(const v4f*)&ker[fbase], ke1 = *(const v4f*)&ker[fbase + 4];

        // ---- invariant filter tile ----
        v8f c = {0, 0, 0, 0, 0, 0, 0, 0};
#pragma unroll
        for (int ks = 0; ks < 9; ++ks)
            c = __builtin_amdgcn_wmma_f32_16x16x4_f32(
                false, afi[ks], false, bfr[ks], (short)0, c, false, false);
#pragma unroll
        for (int r = 0; r < 8; ++r) {
            const float t = (r < 4 ? qi0[r] : qi1[r - 4]) *
                            (r < 4 ? ki0[r] : ki1[r - 4]) * c[r];
            const int f = fbase + r;            // only 'half' is runtime here
            const int h = f / 60;
            ai0 += (h == 0) ? t : 0.f;
            ai1 += (h == 1) ? t : 0.f;
            ai2 += (h == 2) ? t : 0.f;
            ai3 += (h == 3) ? t : 0.f;
        }

        // ---- equivariant filter tile ----
        v8f ce = {0, 0, 0, 0, 0, 0, 0, 0};
#pragma unroll
        for (int ks = 0; ks < 9; ++ks)
            ce = __builtin_amdgcn_wmma_f32_16x16x4_f32(
                false, afe[ks], false, bfr[ks], (short)0, ce, false, false);
#pragma unroll
        for (int r = 0; r < 8; ++r) {
            const float t = (r < 4 ? qe0[r] : qe1[r - 4]) *
                            (r < 4 ? ke0[r] : ke1[r - 4]) * ce[r];
            const int f = fbase + r;
            const int g = f / 80;
            ae0 += (g == 0) ? t : 0.f;
            ae1 += (g == 1) ? t : 0.f;
            ae2 += (g == 2) ? t : 0.f;
        }
    }

    // combine lane-halves (same edge, complementary features)
    ai0 += __shfl_xor(ai0, 16, 32);
    ai1 += __shfl_xor(ai1, 16, 32);
    ai2 += __shfl_xor(ai2, 16, 32);
    ai3 += __shfl_xor(ai3, 16, 32);
    ae0 += __shfl_xor(ae0, 16, 32);
    ae1 += __shfl_xor(ae1, 16, 32);
    ae2 += __shfl_xor(ae2, 16, 32);

    const float si = cut * 0.12909944487358056f;  // 1/sqrt(60)
    const float se = cut * 0.11180339887498949f;  // 1/sqrt(80)
    ai0 *= si; ai1 *= si; ai2 *= si; ai3 *= si;
    ae0 *= se; ae1 *= se; ae2 *= se;

    // scatter-add d_inv: half 0 handles heads 0,1; half 1 handles heads 2,3
#pragma unroll
    for (int hh = 0; hh < 2; ++hh) {
        const float a = half ? (hh ? ai3 : ai2) : (hh ? ai1 : ai0);
        const int h = half * 2 + hh;
        const float* vrow = vi + snd * FDIM + h * 60;
        float* orow = d_inv + rcv * FDIM + h * 60;
#pragma unroll
        for (int qq = 0; qq < 15; ++qq) {
            v4f v = *(const v4f*)&vrow[qq * 4];
#pragma unroll
            for (int u = 0; u < 4; ++u)
                atomicAdd(&orow[qq * 4 + u], a * v[u]);
        }
    }

    // scatter-add d_ev: 15 components, half-0 lane of each edge
    if (half == 0) {
#pragma unroll
        for (int j = 0; j < EV_DIM; ++j) {
            const float a = (j < 3) ? ae0 : (j < 8) ? ae1 : ae2;
            atomicAdd(&d_ev[rcv * EV_DIM + j], a * sh[e * EV_DIM + j]);
        }
    }
}

// --------------------------------------------------------------------------
extern "C" void kernel_launch(void* const* d_in, const int* in_sizes, int n_in,
                              void* d_out, int out_size, void* d_ws, size_t ws_size,
                              hipStream_t stream) {
    const float* inv_features = (const float*)d_in[0];
    const float* ev_features  = (const float*)d_in[1];
    const float* rbf          = (const float*)d_in[2];
    const float* sh           = (const float*)d_in[3];
    const float* cutoffs      = (const float*)d_in[4];
    const float* W_q_inv      = (const float*)d_in[5];
    const float* W_k_inv      = (const float*)d_in[6];
    const float* W_v_inv      = (const float*)d_in[7];
    const float* W_q_ev       = (const float*)d_in[8];
    const float* W_k_ev       = (const float*)d_in[9];
    const float* W_filt_inv   = (const float*)d_in[10];
    const float* b_filt_inv   = (const float*)d_in[11];
    const float* W_filt_ev    = (const float*)d_in[12];
    const float* b_filt_ev    = (const float*)d_in[13];
    const int*   senders      = (const int*)d_in[14];
    const int*   receivers    = (const int*)d_in[15];

    float* ws    = (float*)d_ws;
    float* qi    = ws;                          // [N,240]
    float* ki    = qi + N_NODES * FDIM;
    float* vi    = ki + N_NODES * FDIM;
    float* qe    = vi + N_NODES * FDIM;
    float* ke    = qe + N_NODES * FDIM;
    float* invs  = ke + N_NODES * FDIM;         // [E,3]
    float* wt_qi = invs + N_EDGES * 3;          // [4,64,60]
    float* wt_ki = wt_qi + 4 * 64 * 60;
    float* wt_vi = wt_ki + 4 * 64 * 60;
    float* wt_qe = wt_vi + 4 * 64 * 60;         // [3,80,80]
    float* wt_ke = wt_qe + 3 * 80 * 80;
    float* wtf   = wt_ke + 3 * 80 * 80;         // [2][240][36] inv then ev

    float* d_inv = (float*)d_out;               // [N,240]
    float* d_ev  = d_inv + N_NODES * FDIM;      // [N,15]

    zero_kernel<<<(out_size + 255) / 256, 256, 0, stream>>>((float*)d_out, out_size);

    edge_invs_kernel<<<(N_EDGES + 255) / 256, 256, 0, stream>>>(
        ev_features, senders, receivers, invs);

    // weight pre-transposes (tiny)
    transpose_filt_kernel<<<(WT_SZ + 255) / 256, 256, 0, stream>>>(
        W_filt_inv, b_filt_inv, wtf);
    transpose_filt_kernel<<<(WT_SZ + 255) / 256, 256, 0, stream>>>(
        W_filt_ev, b_filt_ev, wtf + WT_SZ);
    {
        int tot = 4 * 64 * 60;
        transpose_proj_kernel<60, 64><<<(tot + 255) / 256, 256, 0, stream>>>(W_q_inv, wt_qi, tot);
        transpose_proj_kernel<60, 64><<<(tot + 255) / 256, 256, 0, stream>>>(W_k_inv, wt_ki, tot);
        transpose_proj_kernel<60, 64><<<(tot + 255) / 256, 256, 0, stream>>>(W_v_inv, wt_vi, tot);
    }
    {
        int tot = 3 * 80 * 80;
        transpose_proj_kernel<80, 80><<<(tot + 255) / 256, 256, 0, stream>>>(W_q_ev, wt_qe, tot);
        transpose_proj_kernel<80, 80><<<(tot + 255) / 256, 256, 0, stream>>>(W_k_ev, wt_ke, tot);
    }

    // q/k/v invariant projections: 625 node tiles x 4 heads x 4 col tiles
    proj_kernel<60, 64, 3><<<dim3(625, 4, 4), 32, 0, stream>>>(
        inv_features, wt_qi, wt_ki, wt_vi, qi, ki, vi);

    // q/k equivariant projections: 625 node tiles x 3 heads x 5 col tiles
    proj_kernel<80, 80, 2><<<dim3(625, 3, 5), 32, 0, stream>>>(
        inv_features, wt_qe, wt_ke, wt_ke, qe, ke, ke);

    // fused filter GEMM + attention + scatter: 10000 edge tiles, 4 waves/block
    edge_attn_kernel<<<N_EDGES / 16 / 4, 128, 2 * WT_SZ * sizeof(float), stream>>>(
        rbf, sh, cutoffs, wtf,
        senders, receivers, qi, ki, vi, qe, ke, invs, d_inv, d_ev);
}